// ProposalModule_7327214207431
// MI455X (gfx1250) — compile-verified
//
#include <hip/hip_runtime.h>

typedef __attribute__((ext_vector_type(2))) float v2f;
typedef __attribute__((ext_vector_type(8))) float v8f;
typedef __attribute__((ext_vector_type(4))) int   v4i;

#define BATCH 8
#define NPTS  65536
#define CFEAT 256
#define NS    1024
#define CP1   260      // (3+256)=259 padded to multiple of 4
#define OC    128
#define KC    64       // K-chunk staged in LDS per double-buffer step
#define XPITCH 80      // LDS row pitch (floats): 64 + 16 pad -> bank-conflict free

// ---- gfx1250 async global->LDS copy (ASYNCcnt-tracked), with safe fallback --
#if defined(__has_builtin)
# if __has_builtin(__builtin_amdgcn_global_load_async_to_lds_b128)
#  define ASYNC_LDS 1
# endif
#endif
#ifndef ASYNC_LDS
# define ASYNC_LDS 0
#endif

__device__ __forceinline__ void copy16_g2l(const float* g, float* l) {
#if ASYNC_LDS
  __builtin_amdgcn_global_load_async_to_lds_b128(
      (__attribute__((address_space(1))) v4i*)(g),
      (__attribute__((address_space(3))) v4i*)(l), 0, 0);
#else
  float4 t = *(const float4*)g;
  *(float4*)l = t;
#endif
}

__device__ __forceinline__ void async_wait0() {
#if ASYNC_LDS
# if __has_builtin(__builtin_amdgcn_s_wait_asynccnt)
  __builtin_amdgcn_s_wait_asynccnt(0);
# else
  asm volatile("s_wait_asynccnt 0" ::: "memory");
# endif
#endif
}

// Copy kc rows x 64 cols of X (row stride NS) into LDS tile (row pitch XPITCH).
__device__ __forceinline__ void copy_chunk(const float* __restrict__ gsrc,
                                           float* ldst, int kc, int tid) {
  int count = kc * 16;                       // 16 x b128 per 64-float row
  for (int e = tid; e < count; e += 256) {
    int row = e >> 4, c16 = e & 15;
    copy16_g2l(gsrc + (size_t)row * NS + c16 * 4, ldst + row * XPITCH + c16 * 4);
  }
}

// One K-group (K+=4): A pair from LDS weights, 4 B-tiles from LDS X buffer.
__device__ __forceinline__ void wmma_kstep(const float* __restrict__ wrow,
                                           const float* __restrict__ xcur,
                                           int k, int khalf, int mrow, v8f acc[4]) {
  v2f A; A.x = wrow[k]; A.y = wrow[k + 1];
  const float* xk = xcur + (k + khalf) * XPITCH + mrow;
#pragma unroll
  for (int t = 0; t < 4; ++t) {
    v2f Bv; Bv.x = xk[t * 16]; Bv.y = xk[t * 16 + XPITCH];
    acc[t] = __builtin_amdgcn_wmma_f32_16x16x4_f32(false, A, false, Bv,
                                                   (short)0, acc[t], false, false);
  }
}

// ---------------------------------------------------------------------------
// Kernel 1: stable compaction of first NS in-radius indices per batch.
// One block (1024 thr = 32 waves) per batch; ballot + LDS wave-count scan.
// Deterministic (no atomics). Index 0 is always valid (d2==0), so the pad
// value sel[b][0] always exists and equals the reference's order[:,0].
// ---------------------------------------------------------------------------
__global__ void k_select(const float* __restrict__ xyz, int* __restrict__ sel,
                         float* __restrict__ centers) {
  int b = blockIdx.x, tid = threadIdx.x;
  int lane = tid & 31, wv = tid >> 5;
  __shared__ float c3[3];
  __shared__ int wcnt[32];
  __shared__ int s_fill;
  if (tid < 3) c3[tid] = xyz[(size_t)b * NPTS * 3 + tid];
  __syncthreads();
  float cx = c3[0], cy = c3[1], cz = c3[2];
  int base = 0;
  for (int n0 = 0; n0 < NPTS; n0 += 1024) {
    int idx = n0 + tid;
    const float* p = xyz + ((size_t)b * NPTS + idx) * 3;
    float dx = p[0] - cx, dy = p[1] - cy, dz = p[2] - cz;
    bool m = (dx * dx + dy * dy + dz * dz) < 1.0f;   // RADIUS^2 = 1
    unsigned bal = __builtin_amdgcn_ballot_w32(m);
    if (lane == 0) wcnt[wv] = __popc(bal);
    __syncthreads();
    int pre = 0, tot = 0;
    for (int w = 0; w < 32; ++w) { int c = wcnt[w]; if (w < wv) pre += c; tot += c; }
    int pos = base + pre + __popc(bal & ((1u << lane) - 1u));
    if (m && pos < NS) {
      sel[b * NS + pos] = idx;
      if (pos == 0) s_fill = idx;
    }
    base += tot;
    __syncthreads();           // wcnt/s_fill stable before next round / exit
    if (base >= NS) break;     // uniform branch (base identical in all lanes)
  }
  if (base < NS) {
    int fill = s_fill;
    for (int j = base + tid; j < NS; j += 1024) sel[b * NS + j] = fill;
  }
  if (tid < 3) centers[b * 3 + tid] = c3[tid];
}

// ---------------------------------------------------------------------------
// Kernel 2: build G[b][c][n], c in [0,CP1): rows 0..2 = (xyz[ind]-center)/R,
// rows 3..258 = gathered features, row 259 = zero pad (for K%4==0).
// ---------------------------------------------------------------------------
__global__ void k_build_g(const float* __restrict__ xyz, const float* __restrict__ feat,
                          const int* __restrict__ sel, const float* __restrict__ centers,
                          float* __restrict__ G) {
  int blk = blockIdx.x;
  int b = blk / CP1, c = blk % CP1;
  int tid = threadIdx.x;
  float* grow = G + ((size_t)b * CP1 + c) * NS;
  for (int n = tid; n < NS; n += blockDim.x) {
    int ind = sel[b * NS + n];
    float v;
    if (c < 3)        v = xyz[((size_t)b * NPTS + ind) * 3 + c] - centers[b * 3 + c];
    else if (c < 259) v = feat[((size_t)b * CFEAT + (c - 3)) * NPTS + ind];
    else              v = 0.0f;
    grow[n] = v;
  }
}

// ---------------------------------------------------------------------------
// Kernel 3: fp32 WMMA GEMM  Y[b,o,n] = sum_c W[o,c]*X[b,c,n] + bias[o]
//   W (OCxIC row major) staged zero-padded to OCxICP in LDS (coalesced).
//   X K-slabs (KC x 64) double-buffered in LDS via async global->LDS b128
//   copies (ASYNCcnt), copy of chunk i+1 overlapping WMMA of chunk i.
//   256 thr = 8 waves; wave w owns M rows [16w,16w+16), all share 64 N cols.
//   Full-KC chunks run a compile-time-bounded loop (unrolled x4 -> 16 WMMA
//   per trip); only the last ragged chunk uses the runtime-bounded loop.
// A 16x4 layout: lane<16: M=lane, vgprs K{0,1}; lane>=16: M=lane-16, K{2,3}
// B 4x16 layout: lane<16: K=0(+vgpr), N=lane; lane>=16: K=2(+vgpr), N=lane-16
// C/D: vgpr j -> M=j (lanes 0-15) / M=j+8 (lanes 16-31), N=lane&15
// ---------------------------------------------------------------------------
__global__ void k_gemm_wmma(const float* __restrict__ W, const float* __restrict__ bias,
                            const float* __restrict__ X, float* __restrict__ Y,
                            int IC, int ICP) {
  extern __shared__ float lds[];
  float* ldsW  = lds;                         // OC * ICP
  float* xbuf0 = lds + OC * ICP;              // KC * XPITCH
  float* xbuf1 = xbuf0 + KC * XPITCH;         // KC * XPITCH

  int tid = threadIdx.x;
  for (int i = tid; i < OC * ICP; i += 256) {
    int o = i / ICP, c = i - o * ICP;
    ldsW[i] = (c < IC) ? W[o * IC + c] : 0.0f;
  }

  int tilesPerBatch = NS / 64;
  int b     = blockIdx.x / tilesPerBatch;
  int nBase = (blockIdx.x % tilesPerBatch) * 64;
  const float* Xb = X + (size_t)b * ICP * NS + nBase;   // 16B aligned

  int wv = tid >> 5, lane = tid & 31;
  int oBase = wv * 16;
  int mrow  = lane & 15;
  int khalf = (lane < 16) ? 0 : 2;
  int hi8   = (lane < 16) ? 0 : 8;

  v8f acc[4];
#pragma unroll
  for (int j = 0; j < 8; ++j) {
    float bj = bias[oBase + hi8 + j];
    acc[0][j] = bj; acc[1][j] = bj; acc[2][j] = bj; acc[3][j] = bj;
  }

  int nch = (ICP + KC - 1) / KC;
  // prologue: stage chunk 0
  copy_chunk(Xb, xbuf0, (ICP < KC) ? ICP : KC, tid);
  async_wait0();
  __syncthreads();                            // also covers ldsW fill

  for (int ci = 0; ci < nch; ++ci) {
    int k0 = ci * KC;
    int kc = (ICP - k0 < KC) ? (ICP - k0) : KC;
    const float* xcur = (ci & 1) ? xbuf1 : xbuf0;
    if (ci + 1 < nch) {                       // issue async copy of next chunk
      int k1 = k0 + KC;
      int kn = (ICP - k1 < KC) ? (ICP - k1) : KC;
      copy_chunk(Xb + (size_t)k1 * NS, (ci & 1) ? xbuf0 : xbuf1, kn, tid);
    }
    const float* wrow = ldsW + (size_t)(oBase + mrow) * ICP + k0 + khalf;
    if (kc == KC) {
#pragma unroll 4
      for (int k = 0; k < KC; k += 4)
        wmma_kstep(wrow, xcur, k, khalf, mrow, acc);
    } else {
      for (int k = 0; k < kc; k += 4)
        wmma_kstep(wrow, xcur, k, khalf, mrow, acc);
    }
    if (ci + 1 < nch) async_wait0();          // next chunk's copies landed
    __syncthreads();                          // all reads of this buffer done
  }

#pragma unroll
  for (int t = 0; t < 4; ++t) {
    int n = nBase + t * 16 + mrow;
#pragma unroll
    for (int j = 0; j < 8; ++j) {
      int o = oBase + hi8 + j;
      Y[((size_t)b * OC + o) * NS + n] = acc[t][j];
    }
  }
}

// ---------------------------------------------------------------------------
// Kernel 4: per-channel BN stats over (batch, points) -> scale/shift.
// One block per channel, deterministic LDS tree reduction.
// ---------------------------------------------------------------------------
__global__ void k_stats(const float* __restrict__ Y, const float* __restrict__ gamma,
                        const float* __restrict__ beta,
                        float* __restrict__ scale, float* __restrict__ shift) {
  int ch = blockIdx.x, tid = threadIdx.x;
  __shared__ float ss[256], sq[256];
  float s = 0.f, q = 0.f;
  for (int e = tid; e < BATCH * NS; e += 256) {
    int b = e >> 10, n = e & (NS - 1);
    float v = Y[((size_t)b * OC + ch) * NS + n];
    s += v; q += v * v;
  }
  ss[tid] = s; sq[tid] = q; __syncthreads();
  for (int st = 128; st > 0; st >>= 1) {
    if (tid < st) { ss[tid] += ss[tid + st]; sq[tid] += sq[tid + st]; }
    __syncthreads();
  }
  if (tid == 0) {
    const float inv = 1.0f / (BATCH * NS);
    float mean = ss[0] * inv;
    float var  = sq[0] * inv - mean * mean;
    float r    = rsqrtf(var + 1e-5f);
    float sc   = gamma[ch] * r;
    scale[ch] = sc;
    shift[ch] = beta[ch] - mean * sc;
  }
}

// Kernel 5: in-place BN apply + ReLU
__global__ void k_norm(float* __restrict__ Y, const float* __restrict__ scale,
                       const float* __restrict__ shift) {
  int i = blockIdx.x * blockDim.x + threadIdx.x;
  int ch = (i >> 10) & (OC - 1);
  float z = Y[i] * scale[ch] + shift[ch];
  Y[i] = fmaxf(z, 0.0f);
}

// Kernel 6: BN apply + ReLU + max over points.  max(relu(z)) = relu(max(z)).
__global__ void k_maxpool(const float* __restrict__ Y, const float* __restrict__ scale,
                          const float* __restrict__ shift, float* __restrict__ f) {
  int blk = blockIdx.x;                 // b*OC + ch
  int b = blk >> 7, ch = blk & (OC - 1);
  int tid = threadIdx.x;
  __shared__ float sm[256];
  float sc = scale[ch], sh = shift[ch];
  float m = -3.4e38f;
  const float* row = Y + ((size_t)b * OC + ch) * NS;
  for (int n = tid; n < NS; n += 256) m = fmaxf(m, row[n] * sc + sh);
  sm[tid] = m; __syncthreads();
  for (int st = 128; st > 0; st >>= 1) {
    if (tid < st) sm[tid] = fmaxf(sm[tid], sm[tid + st]);
    __syncthreads();
  }
  if (tid == 0) f[blk] = fmaxf(sm[0], 0.0f);
}

// ---------------------------------------------------------------------------
// Kernel 7: head MLP (8x128 @ 128x128 x2 with BN over batch axis, then 12-wide
// output layer). Single block, 128 threads (thread = output channel); BN over
// the 8 batch samples lives entirely in each thread's registers -> exact.
// ---------------------------------------------------------------------------
__global__ void k_head(const float* __restrict__ f,
                       const float* __restrict__ hw1, const float* __restrict__ hb1,
                       const float* __restrict__ hg1, const float* __restrict__ hbe1,
                       const float* __restrict__ hw2, const float* __restrict__ hb2,
                       const float* __restrict__ hg2, const float* __restrict__ hbe2,
                       const float* __restrict__ hw3, const float* __restrict__ hb3,
                       const float* __restrict__ centers, float* __restrict__ out) {
  int o = threadIdx.x;                               // 0..127
  __shared__ float sa[BATCH * OC], sb[BATCH * OC];
  for (int i = o; i < BATCH * OC; i += 128) sa[i] = f[i];
  __syncthreads();

  float v[BATCH];
  // ---- layer 1: sa -> sb ----
  {
#pragma unroll
    for (int bb = 0; bb < BATCH; ++bb) {
      float a = hb1[o];
      for (int c = 0; c < OC; ++c) a += sa[bb * OC + c] * hw1[o * OC + c];
      v[bb] = a;
    }
    float m = 0.f, q = 0.f;
#pragma unroll
    for (int bb = 0; bb < BATCH; ++bb) { m += v[bb]; q += v[bb] * v[bb]; }
    m *= (1.0f / BATCH); q = q * (1.0f / BATCH) - m * m;
    float r = rsqrtf(q + 1e-5f), sc = hg1[o] * r, sh = hbe1[o] - m * sc;
#pragma unroll
    for (int bb = 0; bb < BATCH; ++bb) sb[bb * OC + o] = fmaxf(v[bb] * sc + sh, 0.f);
  }
  __syncthreads();
  // ---- layer 2: sb -> sa ----
  {
#pragma unroll
    for (int bb = 0; bb < BATCH; ++bb) {
      float a = hb2[o];
      for (int c = 0; c < OC; ++c) a += sb[bb * OC + c] * hw2[o * OC + c];
      v[bb] = a;
    }
    float m = 0.f, q = 0.f;
#pragma unroll
    for (int bb = 0; bb < BATCH; ++bb) { m += v[bb]; q += v[bb] * v[bb]; }
    m *= (1.0f / BATCH); q = q * (1.0f / BATCH) - m * m;
    float r = rsqrtf(q + 1e-5f), sc = hg2[o] * r, sh = hbe2[o] - m * sc;
#pragma unroll
    for (int bb = 0; bb < BATCH; ++bb) sa[bb * OC + o] = fmaxf(v[bb] * sc + sh, 0.f);
  }
  __syncthreads();
  // ---- output layer (12 channels): out = [center+net[:3] | size | rot6d] ----
  if (o < 12) {
    for (int bb = 0; bb < BATCH; ++bb) {
      float a = hb3[o];
      for (int c = 0; c < OC; ++c) a += sa[bb * OC + c] * hw3[o * OC + c];
      if (o < 3)      out[bb * 3 + o]             = centers[bb * 3 + o] + a;
      else if (o < 6) out[24 + bb * 3 + (o - 3)]  = a;
      else            out[48 + bb * 6 + (o - 6)]  = a;
    }
  }
}

// ---------------------------------------------------------------------------
extern "C" void kernel_launch(void* const* d_in, const int* in_sizes, int n_in,
                              void* d_out, int out_size, void* d_ws, size_t ws_size,
                              hipStream_t stream) {
  const float* xyz  = (const float*)d_in[0];
  const float* feat = (const float*)d_in[1];
  const float* w1   = (const float*)d_in[2];  const float* b1   = (const float*)d_in[3];
  const float* g1   = (const float*)d_in[4];  const float* be1  = (const float*)d_in[5];
  const float* w2   = (const float*)d_in[6];  const float* b2   = (const float*)d_in[7];
  const float* g2   = (const float*)d_in[8];  const float* be2  = (const float*)d_in[9];
  const float* w3   = (const float*)d_in[10]; const float* b3   = (const float*)d_in[11];
  const float* g3   = (const float*)d_in[12]; const float* be3  = (const float*)d_in[13];
  const float* hw1  = (const float*)d_in[14]; const float* hb1  = (const float*)d_in[15];
  const float* hg1  = (const float*)d_in[16]; const float* hbe1 = (const float*)d_in[17];
  const float* hw2  = (const float*)d_in[18]; const float* hb2  = (const float*)d_in[19];
  const float* hg2  = (const float*)d_in[20]; const float* hbe2 = (const float*)d_in[21];
  const float* hw3  = (const float*)d_in[22]; const float* hb3  = (const float*)d_in[23];
  float* out = (float*)d_out;

  float* ws      = (float*)d_ws;
  float* G       = ws;                                  // 8*260*1024
  float* Y1      = G  + (size_t)BATCH * CP1 * NS;       // 8*128*1024
  float* Y2      = Y1 + (size_t)BATCH * OC * NS;        // 8*128*1024
  float* scale   = Y2 + (size_t)BATCH * OC * NS;        // 128
  float* shift   = scale + OC;                          // 128
  float* fpool   = shift + OC;                          // 1024
  float* centers = fpool + BATCH * OC;                  // 24 (pad to 32)
  int*   sel     = (int*)(centers + 32);                // 8*1024 ints

  k_select<<<BATCH, 1024, 0, stream>>>(xyz, sel, centers);
  k_build_g<<<BATCH * CP1, 256, 0, stream>>>(xyz, feat, sel, centers, G);

  const size_t xlds = (size_t)2 * KC * XPITCH * sizeof(float);      // 40 KB
  const size_t sh1  = (size_t)OC * CP1 * sizeof(float) + xlds;      // ~170 KB
  const size_t sh2  = (size_t)OC * OC  * sizeof(float) + xlds;      // ~104 KB
  const int gemmGrid = (NS / 64) * BATCH;                           // 128 blocks

  k_gemm_wmma<<<gemmGrid, 256, sh1, stream>>>(w1, b1, G,  Y1, 259, CP1);
  k_stats<<<OC, 256, 0, stream>>>(Y1, g1, be1, scale, shift);
  k_norm<<<BATCH * OC * NS / 256, 256, 0, stream>>>(Y1, scale, shift);

  k_gemm_wmma<<<gemmGrid, 256, sh2, stream>>>(w2, b2, Y1, Y2, OC, OC);
  k_stats<<<OC, 256, 0, stream>>>(Y2, g2, be2, scale, shift);
  k_norm<<<BATCH * OC * NS / 256, 256, 0, stream>>>(Y2, scale, shift);

  k_gemm_wmma<<<gemmGrid, 256, sh2, stream>>>(w3, b3, Y2, Y1, OC, OC);
  k_stats<<<OC, 256, 0, stream>>>(Y1, g3, be3, scale, shift);
  k_maxpool<<<BATCH * OC, 256, 0, stream>>>(Y1, scale, shift, fpool);

  k_head<<<1, 128, 0, stream>>>(fpool, hw1, hb1, hg1, hbe1,
                                hw2, hb2, hg2, hbe2, hw3, hb3, centers, out);
}